// Net_80796924772492
// MI455X (gfx1250) — compile-verified
//
#include <hip/hip_runtime.h>

typedef float v2f __attribute__((ext_vector_type(2)));
typedef float v8f __attribute__((ext_vector_type(8)));

// -------------------- utility kernels --------------------

__global__ void fill_zero_f32(float* __restrict__ p, int n) {
  int i = blockIdx.x * blockDim.x + threadIdx.x;
  if (i < n) p[i] = 0.0f;
}

// Copy src[M,K] row-major into zero-padded dst[Mp,Kp].
__global__ void pad2d(const float* __restrict__ src, float* __restrict__ dst,
                      int M, int K, int Mp, int Kp) {
  long idx = (long)blockIdx.x * blockDim.x + threadIdx.x;
  long total = (long)Mp * Kp;
  if (idx >= total) return;
  int c = (int)(idx % Kp);
  int r = (int)(idx / Kp);
  dst[idx] = (r < M && c < K) ? src[(long)r * K + c] : 0.0f;
}

// Replicates build_perm() and scatters grouped 1x1-conv weights w[G, ncpf]
// into dense wd[G, Kp] (row stride Kp, pre-zeroed). One thread per i (i < nf).
__global__ void build_dense_weights(const float* __restrict__ w, float* __restrict__ wd,
                                    int nf, int ncpf, int Kp) {
  int i = threadIdx.x;
  if (i >= nf) return;
  int g = 0;
  for (int ii = 0; ii < i; ++ii)
    g += (nf - ii) + (ncpf - 1) * (nf - 1 - ii);
  for (int j = 0; j < ncpf; ++j) {
    if (j == 0) {
      for (int k = i; k < nf; ++k) {
        wd[(long)g * Kp + i * ncpf] = w[(long)g * ncpf + 0];
        for (int t = 1; t < ncpf; ++t)
          wd[(long)g * Kp + k * ncpf + t] = w[(long)g * ncpf + t];
        ++g;
      }
    } else {
      for (int k = i + 1; k < nf; ++k) {
        for (int t = 0; t < ncpf; ++t) {
          int ch = (t == j) ? (i * ncpf + j) : (k * ncpf + t);
          wd[(long)g * Kp + ch] = w[(long)g * ncpf + t];
        }
        ++g;
      }
    }
  }
}

// Grouped 5x5 VALID conv writing a zero-padded GEMM-B layout:
// out[b][Coutp rows x Np cols]; row oc in [0,Cout) col p in [0,Hout*Wout) hold
// conv results, everything else written as 0.
__global__ void dwconv5x5_pad(const float* __restrict__ x, const float* __restrict__ w,
                              const float* __restrict__ bias, float* __restrict__ out,
                              int B, int Cin, int Cout, int Hin, int Win, int mult,
                              int Coutp, int Np) {
  int Hout = Hin - 4, Wout = Win - 4;
  long total = (long)B * Coutp * Np;
  long idx = (long)blockIdx.x * blockDim.x + threadIdx.x;
  if (idx >= total) return;
  int p = (int)(idx % Np);
  long t = idx / Np;
  int oc = (int)(t % Coutp);
  int b = (int)(t / Coutp);
  if (oc >= Cout || p >= Hout * Wout) { out[idx] = 0.0f; return; }
  int yh = p / Wout, xw = p % Wout;
  int ic = oc / mult;
  const float* xp = x + (((long)b * Cin + ic) * Hin + yh) * Win + xw;
  const float* wp = w + (long)oc * 25;
  float acc = bias[oc];
#pragma unroll
  for (int dy = 0; dy < 5; ++dy)
#pragma unroll
    for (int dx = 0; dx < 5; ++dx)
      acc = fmaf(xp[dy * Win + dx], wp[dy * 5 + dx], acc);
  out[idx] = acc;
}

// relu+2x2 maxpool, compact output [BC, Ho*Wo]
__global__ void relu_maxpool2(const float* __restrict__ in, float* __restrict__ out,
                              int BC, int Hin, int Win) {
  int Ho = Hin >> 1, Wo = Win >> 1;
  long total = (long)BC * Ho * Wo;
  long idx = (long)blockIdx.x * blockDim.x + threadIdx.x;
  if (idx >= total) return;
  int xo = (int)(idx % Wo); long t = idx / Wo;
  int yo = (int)(t % Ho); int bc = (int)(t / Ho);
  const float* p = in + (((long)bc * Hin) + 2 * yo) * Win + 2 * xo;
  float m = fmaxf(fmaxf(p[0], p[1]), fmaxf(p[Win], p[Win + 1]));
  out[idx] = fmaxf(m, 0.0f);
}

// relu+2x2 maxpool writing a zero-padded GEMM-B layout [Cp rows x Np cols] per batch.
// Input is compact [B][C, Hin*Win] with batch stride inBatch.
__global__ void relu_maxpool2_pad(const float* __restrict__ in, float* __restrict__ out,
                                  int B, int C, int Hin, int Win, int Cp, int Np,
                                  long inBatch) {
  int Ho = Hin >> 1, Wo = Win >> 1;
  long total = (long)B * Cp * Np;
  long idx = (long)blockIdx.x * blockDim.x + threadIdx.x;
  if (idx >= total) return;
  int pcol = (int)(idx % Np);
  long t = idx / Np;
  int c = (int)(t % Cp);
  int b = (int)(t / Cp);
  if (c >= C || pcol >= Ho * Wo) { out[idx] = 0.0f; return; }
  int yo = pcol / Wo, xo = pcol % Wo;
  const float* p = in + (long)b * inBatch + ((long)c * Hin + 2 * yo) * Win + 2 * xo;
  float m = fmaxf(fmaxf(p[0], p[1]), fmaxf(p[Win], p[Win + 1]));
  out[idx] = fmaxf(m, 0.0f);
}

// -------------------- fp32 WMMA GEMM (padded operands, no inner guards) ---------
// C[M,N] = A[Mp,Kp](zero-padded) * B + bias[M], optional ReLU, batched over grid.z.
// BT=false: B is [Kp rows x Np cols], zero-padded.  BT=true: Bt is [Np rows x Kp].
// One wave per 16x16 tile; 4 waves/block along N. Inner loop: unconditional
// float2 A load + B loads + v_wmma_f32_16x16x4_f32. Guards only in the epilogue.
template <bool BT, bool RELU>
__global__ __launch_bounds__(128) void gemm_wmma_f32(
    const float* __restrict__ A, const float* __restrict__ Bm,
    const float* __restrict__ bias, float* __restrict__ C,
    int M, int N, int Kp, int Np, long long strideB, long long strideC) {
  const int wid = threadIdx.x >> 5;
  const int lane = threadIdx.x & 31;
  const int tn = (blockIdx.x * 4 + wid) * 16;
  if (tn >= N) return;  // wave-uniform: EXEC stays all-ones for live waves
  const int tm = blockIdx.y * 16;

  const int h = lane >> 4;  // selects K pair {2h, 2h+1}
  const int l = lane & 15;
  const int m = tm + l;  // always < Mp (A padded)
  const int n = tn + l;  // always < Np (B padded)

  const float* Ap = A + (long)m * Kp + 2 * h;
  const float* Bp;
  if (BT)
    Bp = Bm + (long long)blockIdx.z * strideB + (long)n * Kp + 2 * h;
  else
    Bp = Bm + (long long)blockIdx.z * strideB + (long)(2 * h) * Np + n;

  v8f acc = {};
#pragma unroll 4
  for (int k0 = 0; k0 < Kp; k0 += 4) {
    v2f av = *(const v2f*)Ap;
    v2f bv;
    if (BT) {
      bv = *(const v2f*)Bp;
    } else {
      bv.x = Bp[0];
      bv.y = Bp[Np];
    }
    acc = __builtin_amdgcn_wmma_f32_16x16x4_f32(
        /*neg_a=*/false, av, /*neg_b=*/false, bv,
        /*c_mod=*/(short)0, acc, /*reuse_a=*/false, /*reuse_b=*/false);
    Ap += 4;
    Bp += BT ? 4 : 4 * (long)Np;
  }

  float* Cp = C + (long long)blockIdx.z * strideC;
  const bool nok = (n < N);
#pragma unroll
  for (int v = 0; v < 8; ++v) {
    int mm = tm + 8 * h + v;
    if (mm < M && nok) {
      float val = acc[v] + bias[mm];
      if (RELU) val = fmaxf(val, 0.0f);
      Cp[(long)mm * N + n] = val;
    }
  }
}

// Tiny final FC: out[b*O+o] = bias[o] + sum_c w[o,c] * h[c, b]   (h is [C, Bn])
__global__ void fc_small(const float* __restrict__ w, const float* __restrict__ bias,
                         const float* __restrict__ h, float* __restrict__ out,
                         int Bn, int O, int C) {
  int idx = blockIdx.x * blockDim.x + threadIdx.x;
  if (idx >= Bn * O) return;
  int o = idx % O, b = idx / O;
  float acc = bias[o];
  for (int c = 0; c < C; ++c)
    acc = fmaf(w[(long)o * C + c], h[(long)c * Bn + b], acc);
  out[idx] = acc;
}

// -------------------- host-side helpers --------------------

template <bool BT, bool RELU>
static inline void launch_gemm(const float* A, const float* Bm, const float* bias,
                               float* C, int M, int N, int Kp, int Np, long long sB,
                               long long sC, int batch, hipStream_t stream) {
  int ntiles = (N + 15) / 16;
  dim3 grid((ntiles + 3) / 4, (M + 15) / 16, batch);
  gemm_wmma_f32<BT, RELU><<<grid, 128, 0, stream>>>(A, Bm, bias, C, M, N, Kp, Np, sB, sC);
}

static inline unsigned gblk(long total) { return (unsigned)((total + 255) / 256); }

extern "C" void kernel_launch(void* const* d_in, const int* in_sizes, int n_in,
                              void* d_out, int out_size, void* d_ws, size_t ws_size,
                              hipStream_t stream) {
  (void)in_sizes; (void)n_in; (void)out_size; (void)ws_size;
  const float* x     = (const float*)d_in[0];
  const float* dw1_w = (const float*)d_in[1];
  const float* dw1_b = (const float*)d_in[2];
  const float* pc1_w = (const float*)d_in[3];
  const float* pc1_b = (const float*)d_in[4];
  const float* dw2_w = (const float*)d_in[5];
  const float* dw2_b = (const float*)d_in[6];
  const float* pc2_w = (const float*)d_in[7];
  const float* pc2_b = (const float*)d_in[8];
  const float* oo_w  = (const float*)d_in[9];
  const float* oo_b  = (const float*)d_in[10];
  const float* fc1_w = (const float*)d_in[11];
  const float* fc1_b = (const float*)d_in[12];
  const float* fc2_w = (const float*)d_in[13];
  const float* fc2_b = (const float*)d_in[14];
  const float* fc3_w = (const float*)d_in[15];
  const float* fc3_b = (const float*)d_in[16];
  float* out = (float*)d_out;
  float* ws  = (float*)d_ws;

  const int B = 64;

  // Padded GEMM shapes:
  //  pc1: M=51 (Mp 64),  Kp=20,   N=784 (Np 784)
  //  pc2: M=515 (Mp 528),Kp=256,  N=100 (Np 112)
  //  oo : M=100 (Mp 112),Kp=516,  N=25  (Np 32)
  //  fc1: M=120 (Mp 128),Kp=2500, N=64  (Np 64, BT)
  //  fc2: M=84  (Mp 96), Kp=120,  N=64  (Np 64)

  // Workspace layout (floats, offsets multiple of 1024 => 4KB aligned)
  float* wd1   = ws;            // 64*20      = 1280
  float* wd2   = ws + 2048;     // 528*256    = 135168
  float* ooWp  = ws + 137216;   // 112*516    = 57792
  float* fc1Wp = ws + 196608;   // 128*2500   = 320000
  float* fc2Wp = ws + 517120;   // 96*120     = 11520
  float* bufA  = ws + 529408;   // max 515*100*64 = 3,296,000
  float* bufB  = ws + 3826688;  // max 51*784*64  = 2,558,976
  // total ~6.39M floats (~25.5 MB)

  // 0) stage padded weight matrices
  fill_zero_f32<<<gblk(64 * 20), 256, 0, stream>>>(wd1, 64 * 20);
  fill_zero_f32<<<gblk(528 * 256), 256, 0, stream>>>(wd2, 528 * 256);
  build_dense_weights<<<1, 32, 0, stream>>>(pc1_w, wd1, 6, 3, 20);
  build_dense_weights<<<1, 32, 0, stream>>>(pc2_w, wd2, 5, 51, 256);
  pad2d<<<gblk((long)112 * 516), 256, 0, stream>>>(oo_w, ooWp, 100, 515, 112, 516);
  pad2d<<<gblk((long)128 * 2500), 256, 0, stream>>>(fc1_w, fc1Wp, 120, 2500, 128, 2500);
  pad2d<<<gblk((long)96 * 120), 256, 0, stream>>>(fc2_w, fc2Wp, 84, 120, 96, 120);

  // 1) dw1: x[64,3,32,32] -> bufA padded [64][20 x 784]
  dwconv5x5_pad<<<gblk((long)B * 20 * 784), 256, 0, stream>>>(
      x, dw1_w, dw1_b, bufA, B, 3, 18, 32, 32, 6, 20, 784);

  // 2) pc1 GEMM: wd1[64,20] x bufA[b][20,784] -> bufB[b][51,784] (compact)
  launch_gemm<false, false>(wd1, bufA, pc1_b, bufB, 51, 784, 20, 784,
                            (long long)20 * 784, (long long)51 * 784, B, stream);

  // 3) relu+pool: bufB [64*51,28,28] -> bufA compact [64,51,14,14]
  relu_maxpool2<<<gblk((long)B * 51 * 14 * 14), 256, 0, stream>>>(bufB, bufA, B * 51, 28, 28);

  // 4) dw2: bufA[64,51,14,14] -> bufB padded [64][256 x 112]
  dwconv5x5_pad<<<gblk((long)B * 256 * 112), 256, 0, stream>>>(
      bufA, dw2_w, dw2_b, bufB, B, 51, 255, 14, 14, 5, 256, 112);

  // 5) pc2 GEMM: wd2[528,256] x bufB[b][256,112] -> bufA[b][515,100] (compact)
  launch_gemm<false, false>(wd2, bufB, pc2_b, bufA, 515, 100, 256, 112,
                            (long long)256 * 112, (long long)515 * 100, B, stream);

  // 6) relu+pool: bufA [64][515,10,10] -> bufB padded [64][516 x 32]
  relu_maxpool2_pad<<<gblk((long)B * 516 * 32), 256, 0, stream>>>(
      bufA, bufB, B, 515, 10, 10, 516, 32, (long)515 * 100);

  // 7) oo GEMM + ReLU: ooWp[112,516] x bufB[b][516,32] -> bufA[b][100,25] == h5 [64,2500]
  launch_gemm<false, true>(ooWp, bufB, oo_b, bufA, 100, 25, 516, 32,
                           (long long)516 * 32, (long long)2500, B, stream);

  // 8) fc1 + ReLU (BT): fc1Wp[128,2500] x h5^T (bufA as [64,2500]) -> bufB [120,64]
  launch_gemm<true, true>(fc1Wp, bufA, fc1_b, bufB, 120, 64, 2500, 64, 0, 0, 1, stream);

  // 9) fc2 + ReLU: fc2Wp[96,120] x bufB[120,64] -> bufA [84,64]
  launch_gemm<false, true>(fc2Wp, bufB, fc2_b, bufA, 84, 64, 120, 64, 0, 0, 1, stream);

  // 10) fc3 -> out [64,10]
  fc_small<<<gblk(64 * 10), 256, 0, stream>>>(fc3_w, fc3_b, bufA, out, 64, 10, 84);
}